// TemporalGCNGraphEmbedding_het_70875550319264
// MI455X (gfx1250) — compile-verified
//
#include <hip/hip_runtime.h>
#include <hip/hip_bf16.h>

// Problem constants (match reference)
#define NN   20000
#define EE   100000
#define TT   3
#define GG   32
#define FIND 16
#define HHD  32
#define OUTD 64
#define FED  8

typedef __attribute__((ext_vector_type(2))) float v2f;
typedef __attribute__((ext_vector_type(8))) float v8f;

// ---------------------------------------------------------------------------
// Utility kernels
// ---------------------------------------------------------------------------
__global__ void zero_kernel(float* __restrict__ p, int n) {
  int i = blockIdx.x * blockDim.x + threadIdx.x;
  if (i < n) p[i] = 0.0f;
}

// xb1[t,n,o] = sum_i x_t[t,n,i] * conv_b2[t, i*32+o]   (b2 contribution per source node)
__global__ void xb1_kernel(const float* __restrict__ x_t,
                           const float* __restrict__ conv_b2,
                           float* __restrict__ xb1) {
  int id = blockIdx.x * blockDim.x + threadIdx.x;
  if (id >= TT * NN) return;
  int t = id / NN, n = id % NN;
  const float* xr = x_t + ((long)t * NN + n) * FIND;
  float xi[FIND];
#pragma unroll
  for (int i = 0; i < FIND; ++i) xi[i] = xr[i];
  const float* b2 = conv_b2 + (long)t * FIND * HHD;
  float* o = xb1 + ((long)t * NN + n) * HHD;
#pragma unroll
  for (int oo = 0; oo < HHD; ++oo) {
    float acc = 0.0f;
#pragma unroll
    for (int i = 0; i < FIND; ++i) acc += xi[i] * b2[i * HHD + oo];
    o[oo] = acc;
  }
}

// in-degree per (t, node)
__global__ void cnt_kernel(const int* __restrict__ eidx, float* __restrict__ cnt) {
  int id = blockIdx.x * blockDim.x + threadIdx.x;
  if (id >= TT * EE) return;
  int t = id / EE, e = id % EE;
  int d = eidx[(long)t * 2 * EE + EE + e];
  unsafeAtomicAdd(cnt + (long)t * NN + d, 1.0f);
}

// nodes per graph
__global__ void gcnt_kernel(const int* __restrict__ batch, float* __restrict__ gcnt) {
  int n = blockIdx.x * blockDim.x + threadIdx.x;
  if (n >= NN) return;
  unsafeAtomicAdd(gcnt + batch[n], 1.0f);
}

// ---------------------------------------------------------------------------
// Core WMMA kernel: fused edge-MLP + per-edge matmul + scatter-add.
//   msg[e,o] = sum_{i,k} (x[src[e],i] * h[e,k]) * w2[k, i*32+o]  (+ x[src]@b2 via xb)
// One wave handles a 16-edge tile; 16 waves (512 thr) share one LDS w2 copy.
// A built in registers (v_pk_mul outer product, hidden kept as v2f pairs);
// B staged in LDS in a permuted layout so each lane's fragment pair is one
// aligned ds_load_2addr_b64.
//   LDS layout: idx = (i*8 + kc)*128 + half*64 + col*2 + v
//               holds w2[kc*4 + 2*half + v, i*32 + col]
// ---------------------------------------------------------------------------
template <int CIN>
__global__ __launch_bounds__(512) void msg_kernel(
    const float* __restrict__ xin,  // [T,N,CIN]
    const float* __restrict__ eat,  // [T,E,8]
    const int* __restrict__ eidx,   // [T,2,E]
    const float* __restrict__ w1,   // [T?,8,32]
    const float* __restrict__ b1,   // [T?,32]
    const float* __restrict__ w2,   // [T?,32,CIN*32]
    const float* __restrict__ xb,   // [T,N,32]
    float* __restrict__ agg,        // [T,N,32]
    int perT) {
  extern __shared__ float w2_lds[];  // HHD * CIN * HHD floats, permuted
  const int t = blockIdx.y;
  const int tid = threadIdx.x;
  const int lane = tid & 31;
  const int wv = tid >> 5;
  const int half = lane >> 4;  // which 16-lane half
  const int n16 = lane & 15;
  const int W2COLS = CIN * HHD;
  const int W2SZ = HHD * W2COLS;

  // cooperative stage of w2[t] into LDS with B-fragment-friendly permutation
  const float* w2t = w2 + (perT ? (long)t * W2SZ : 0);
  for (int idx = tid * 4; idx < W2SZ; idx += 512 * 4) {
    float4 v = *reinterpret_cast<const float4*>(w2t + idx);
    const int k = idx / W2COLS;       // 4 consecutive floats share k (32|W2COLS)
    const int r = idx - k * W2COLS;
    const int ii = r >> 5;
    const int col = r & 31;           // col..col+3 within same 32-block
    const int kcs = k >> 2;
    const int hlf = (k >> 1) & 1;
    const int vv = k & 1;
    const int d = (ii * 8 + kcs) * 128 + hlf * 64 + col * 2 + vv;
    w2_lds[d] = v.x;
    w2_lds[d + 2] = v.y;
    w2_lds[d + 4] = v.z;
    w2_lds[d + 6] = v.w;
  }
  __syncthreads();

  const int NTILES = (EE + 15) / 16;
  const int tile = blockIdx.x * 16 + wv;
  if (tile >= NTILES) return;  // wave-uniform; no barriers after this point
  const int e0 = tile * 16;

  const int* srcp = eidx + (long)t * 2 * EE;
  const int* dstp = srcp + EE;

  const int eA = e0 + n16;  // edge for A-row m = n16 (both halves)
  const int eAc = (eA < EE) ? eA : (EE - 1);
  const float valid = (eA < EE) ? 1.0f : 0.0f;
  const int sA = srcp[eAc];

  // preload the whole x row into registers (vector global loads, waits once)
  const float* xrow = xin + ((long)t * NN + sA) * CIN;
  float xr[CIN];
#pragma unroll
  for (int i = 0; i < CIN; ++i) xr[i] = xrow[i];

  // edge features for this row
  const float* earow = eat + ((long)t * EE + eAc) * FED;
  float ear[FED];
#pragma unroll
  for (int c = 0; c < FED; ++c) ear[c] = earow[c];

  // hidden fragments as aligned pairs: hr2[kc] = h[m, kc*4+2*half + {0,1}]
  const float* w1t = w1 + (perT ? (long)t * FED * HHD : 0);
  const float* b1t = b1 + (perT ? (long)t * HHD : 0);
  v2f hr2[8];
#pragma unroll
  for (int kc = 0; kc < 8; ++kc) {
    const int k0 = kc * 4 + 2 * half;
    float a0 = b1t[k0];
    float a1 = b1t[k0 + 1];
#pragma unroll
    for (int c = 0; c < FED; ++c) {
      a0 += ear[c] * w1t[c * HHD + k0];
      a1 += ear[c] * w1t[c * HHD + k0 + 1];
    }
    hr2[kc].x = fmaxf(a0, 0.0f) * valid;
    hr2[kc].y = fmaxf(a1, 0.0f) * valid;
  }

  // K loop: K = CIN*32, stepped 4 at a time; fully unrolled straight-line code
  v8f c_lo = {};  // outputs o = 0..15
  v8f c_hi = {};  // outputs o = 16..31
  const float* Blane = w2_lds + half * 64 + (n16 << 1);
#pragma unroll
  for (int i = 0; i < CIN; ++i) {
    const float xv = xr[i];
#pragma unroll
    for (int kc = 0; kc < 8; ++kc) {
      v2f a;
      a.x = hr2[kc].x * xv;  // pairs -> single v_pk_mul_f32
      a.y = hr2[kc].y * xv;
      const float* Bb = Blane + (i * 8 + kc) * 128;
      v2f blo = *reinterpret_cast<const v2f*>(Bb);
      v2f bhi = *reinterpret_cast<const v2f*>(Bb + 32);
      c_lo = __builtin_amdgcn_wmma_f32_16x16x4_f32(false, a, false, blo,
                                                   (short)0, c_lo, false, false);
      c_hi = __builtin_amdgcn_wmma_f32_16x16x4_f32(false, a, false, bhi,
                                                   (short)0, c_hi, false, false);
    }
  }

  // scatter-add: C element (M = r + 8*half, N = n16 [+16])
  const float* xbt = xb + (long)t * NN * HHD;
  float* aggt = agg + (long)t * NN * HHD;
#pragma unroll
  for (int r = 0; r < 8; ++r) {
    const int e = e0 + r + 8 * half;
    if (e < EE) {
      const int s = srcp[e];
      const int d = dstp[e];
      const float vlo = c_lo[r] + xbt[(long)s * HHD + n16];
      const float vhi = c_hi[r] + xbt[(long)s * HHD + 16 + n16];
      unsafeAtomicAdd(aggt + (long)d * HHD + n16, vlo);
      unsafeAtomicAdd(aggt + (long)d * HHD + 16 + n16, vhi);
    }
  }
}

// ---------------------------------------------------------------------------
// x1 = relu(agg1/cnt + x@root + bias);  xb2 = x1 @ c2_b2_reshaped
// ---------------------------------------------------------------------------
__global__ void finish1_kernel(const float* __restrict__ x_t,
                               const float* __restrict__ agg1,
                               const float* __restrict__ cnt,
                               const float* __restrict__ root,   // [T,16,32]
                               const float* __restrict__ bias,   // [T,32]
                               const float* __restrict__ c2_b2,  // [1024]
                               float* __restrict__ x1,
                               float* __restrict__ xb2) {
  int id = blockIdx.x * blockDim.x + threadIdx.x;
  if (id >= TT * NN) return;
  int t = id / NN, n = id % NN;
  float cr = cnt[(long)t * NN + n];
  cr = cr < 1.0f ? 1.0f : cr;
  const float* xr = x_t + ((long)t * NN + n) * FIND;
  float xi[FIND];
#pragma unroll
  for (int i = 0; i < FIND; ++i) xi[i] = xr[i];
  const float* rt = root + (long)t * FIND * HHD;
  const float* ag = agg1 + ((long)t * NN + n) * HHD;
  float xo[HHD];
#pragma unroll
  for (int o = 0; o < HHD; ++o) {
    float acc = bias[(long)t * HHD + o] + ag[o] / cr;
#pragma unroll
    for (int i = 0; i < FIND; ++i) acc += xi[i] * rt[i * HHD + o];
    xo[o] = fmaxf(acc, 0.0f);
  }
  float* x1p = x1 + ((long)t * NN + n) * HHD;
#pragma unroll
  for (int o = 0; o < HHD; ++o) x1p[o] = xo[o];
  float* xb2p = xb2 + ((long)t * NN + n) * HHD;
#pragma unroll
  for (int o = 0; o < HHD; ++o) {
    float acc = 0.0f;
#pragma unroll
    for (int i = 0; i < HHD; ++i) acc += xo[i] * c2_b2[i * HHD + o];
    xb2p[o] = acc;
  }
}

// ---------------------------------------------------------------------------
// x2 = agg2/cnt + x1@c2_root + c2_bias;  pooled[g,t,:] += x2 / gcnt[g]
// ---------------------------------------------------------------------------
__global__ void finish2_kernel(const float* __restrict__ x1,
                               const float* __restrict__ agg2,
                               const float* __restrict__ cnt,
                               const float* __restrict__ c2_root,  // [32,32]
                               const float* __restrict__ c2_bias,  // [32]
                               const int* __restrict__ batch,
                               const float* __restrict__ gcnt,
                               float* __restrict__ pooled) {  // [G,T,32]
  int id = blockIdx.x * blockDim.x + threadIdx.x;
  if (id >= TT * NN) return;
  int t = id / NN, n = id % NN;
  float cr = cnt[(long)t * NN + n];
  cr = cr < 1.0f ? 1.0f : cr;
  int g = batch[n];
  float gc = gcnt[g];
  gc = gc < 1.0f ? 1.0f : gc;
  const float* xr = x1 + ((long)t * NN + n) * HHD;
  float xi[HHD];
#pragma unroll
  for (int i = 0; i < HHD; ++i) xi[i] = xr[i];
  const float* ag = agg2 + ((long)t * NN + n) * HHD;
#pragma unroll
  for (int o = 0; o < HHD; ++o) {
    float acc = c2_bias[o] + ag[o] / cr;
#pragma unroll
    for (int i = 0; i < HHD; ++i) acc += xi[i] * c2_root[i * HHD + o];
    unsafeAtomicAdd(pooled + ((long)g * TT + t) * HHD + o, acc / gc);
  }
}

// out[g,t,u] = pooled[g,t,:] @ lin_w + lin_b
__global__ void head_kernel(const float* __restrict__ pooled,
                            const float* __restrict__ lin_w,
                            const float* __restrict__ lin_b,
                            float* __restrict__ out) {
  int id = blockIdx.x * blockDim.x + threadIdx.x;
  if (id >= GG * TT * OUTD) return;
  int g = id / (TT * OUTD);
  int r = id % (TT * OUTD);
  int t = r / OUTD, u = r % OUTD;
  const float* p = pooled + ((long)g * TT + t) * HHD;
  float acc = lin_b[u];
#pragma unroll
  for (int o = 0; o < HHD; ++o) acc += p[o] * lin_w[o * OUTD + u];
  out[id] = acc;
}

// ---------------------------------------------------------------------------
extern "C" void kernel_launch(void* const* d_in, const int* in_sizes, int n_in,
                              void* d_out, int out_size, void* d_ws, size_t ws_size,
                              hipStream_t stream) {
  const float* x_t = (const float*)d_in[0];
  const float* eat = (const float*)d_in[1];
  const int* eidx = (const int*)d_in[2];
  const int* batch = (const int*)d_in[3];
  const float* conv_w1 = (const float*)d_in[4];
  const float* conv_b1 = (const float*)d_in[5];
  const float* conv_w2 = (const float*)d_in[6];
  const float* conv_b2 = (const float*)d_in[7];
  const float* conv_root = (const float*)d_in[8];
  const float* conv_bias = (const float*)d_in[9];
  const float* c2_w1 = (const float*)d_in[10];
  const float* c2_b1 = (const float*)d_in[11];
  const float* c2_w2 = (const float*)d_in[12];
  const float* c2_b2 = (const float*)d_in[13];
  const float* c2_root = (const float*)d_in[14];
  const float* c2_bias = (const float*)d_in[15];
  const float* lin_w = (const float*)d_in[16];
  const float* lin_b = (const float*)d_in[17];

  float* ws = (float*)d_ws;
  const long TN32 = (long)TT * NN * HHD;
  float* agg1 = ws;
  float* agg2 = agg1 + TN32;
  float* cnt = agg2 + TN32;
  float* pooled = cnt + (long)TT * NN;
  float* gcnt = pooled + (long)GG * TT * HHD;
  float* xb1 = gcnt + GG;
  float* xb2 = xb1 + TN32;
  float* x1 = xb2 + TN32;

  // zero the atomically-accumulated regions (contiguous prefix)
  int zn = (int)(2 * TN32 + (long)TT * NN + (long)GG * TT * HHD + GG);
  zero_kernel<<<(zn + 255) / 256, 256, 0, stream>>>(agg1, zn);

  xb1_kernel<<<(TT * NN + 255) / 256, 256, 0, stream>>>(x_t, conv_b2, xb1);
  cnt_kernel<<<(TT * EE + 255) / 256, 256, 0, stream>>>(eidx, cnt);
  gcnt_kernel<<<(NN + 255) / 256, 256, 0, stream>>>(batch, gcnt);

  const int NTILES = (EE + 15) / 16;
  dim3 mgrid((NTILES + 15) / 16, TT);
  const int lds1 = HHD * FIND * HHD * 4;  // 64 KB
  const int lds2 = HHD * HHD * HHD * 4;   // 128 KB (CDNA5 WGP: 320 KB)
  hipFuncSetAttribute((const void*)msg_kernel<FIND>,
                      hipFuncAttributeMaxDynamicSharedMemorySize, lds1);
  hipFuncSetAttribute((const void*)msg_kernel<HHD>,
                      hipFuncAttributeMaxDynamicSharedMemorySize, lds2);

  msg_kernel<FIND><<<mgrid, 512, lds1, stream>>>(x_t, eat, eidx, conv_w1, conv_b1,
                                                 conv_w2, xb1, agg1, 1);
  finish1_kernel<<<(TT * NN + 255) / 256, 256, 0, stream>>>(
      x_t, agg1, cnt, conv_root, conv_bias, c2_b2, x1, xb2);
  msg_kernel<HHD><<<mgrid, 512, lds2, stream>>>(x1, eat, eidx, c2_w1, c2_b1,
                                                c2_w2, xb2, agg2, 0);
  finish2_kernel<<<(TT * NN + 255) / 256, 256, 0, stream>>>(
      x1, agg2, cnt, c2_root, c2_bias, batch, gcnt, pooled);
  head_kernel<<<(GG * TT * OUTD + 255) / 256, 256, 0, stream>>>(
      pooled, lin_w, lin_b, (float*)d_out);
}